// AdaptivePriorBoxesLoss_29076928594015
// MI455X (gfx1250) — compile-verified
//
#include <hip/hip_runtime.h>

typedef float v2f __attribute__((ext_vector_type(2)));
typedef float v8f __attribute__((ext_vector_type(8)));
typedef int   gv4i __attribute__((vector_size(16)));   // matches builtin's V4i param

#define TMAX 256

#define HAS_ASYNC_LDS __has_builtin(__builtin_amdgcn_global_load_async_to_lds_b128)

__device__ __forceinline__ void wait_async_lds() {
#if __has_builtin(__builtin_amdgcn_s_wait_asynccnt)
    __builtin_amdgcn_s_wait_asynccnt(0);
#else
    asm volatile("s_wait_asynccnt 0x0" ::: "memory");
#endif
}

// Stage T truth boxes ([T,4] f32, 16B each) into LDS.
// CDNA5 path: GLOBAL_LOAD_ASYNC_TO_LDS_B128 (ASYNCcnt-tracked DMA into LDS).
__device__ __forceinline__ void stage_truths(const float* __restrict__ truths,
                                             float* tbox, int T, int tid) {
#if HAS_ASYNC_LDS
    if (tid < T) {
        __attribute__((address_space(1))) gv4i* g =
            (__attribute__((address_space(1))) gv4i*)(truths + (size_t)tid * 4);
        __attribute__((address_space(3))) gv4i* l =
            (__attribute__((address_space(3))) gv4i*)(tbox + (size_t)tid * 4);
        __builtin_amdgcn_global_load_async_to_lds_b128(g, l, 0, 0);
    }
    wait_async_lds();
#else
    for (int t = tid; t < T; t += 256) {
        tbox[t * 4 + 0] = truths[t * 4 + 0];
        tbox[t * 4 + 1] = truths[t * 4 + 1];
        tbox[t * 4 + 2] = truths[t * 4 + 2];
        tbox[t * 4 + 3] = truths[t * 4 + 3];
    }
#endif
}

__device__ __forceinline__ unsigned long long shfl_xor_u64(unsigned long long v, int mask) {
    unsigned lo = (unsigned)(v & 0xffffffffull);
    unsigned hi = (unsigned)(v >> 32);
    lo = __shfl_xor(lo, mask, 32);
    hi = __shfl_xor(hi, mask, 32);
    return ((unsigned long long)hi << 32) | (unsigned long long)lo;
}

// Phase 1: per-prior best-truth (max/argmax over T) + per-truth best-prior
// (global argmax via packed u64 atomicMax; wave32 shfl reduce -> LDS -> 1 atomic per (block,t)).
__global__ __launch_bounds__(256) void apl_phase1(
    const float* __restrict__ locs, const float* __restrict__ params,
    const float* __restrict__ truths, int P, int T,
    unsigned long long* __restrict__ packed,
    float* __restrict__ bto, int* __restrict__ bti)
{
    __shared__ __align__(16) float tbox[TMAX * 4];
    __shared__ unsigned long long wmax[TMAX * 8];

    int tid = threadIdx.x;
    stage_truths(truths, tbox, T, tid);
    __syncthreads();

    int p = blockIdx.x * 256 + tid;
    if (p >= P) p = P - 1;              // clamp: keeps wave uniform; duplicate work is idempotent

    float lx = locs[2 * p + 0], ly = locs[2 * p + 1];
    float w  = params[3 * p + 0], h = params[3 * p + 1];
    float px0 = lx - 0.5f * w, py0 = ly - 0.5f * h;
    float px1 = lx + 0.5f * w, py1 = ly + 0.5f * h;
    float areaB = w * h;

    float best = -1.0f; int bidx = 0;
    int lane = tid & 31, wv = tid >> 5;

    for (int t = 0; t < T; ++t) {
        float ax = tbox[t * 4 + 0], ay = tbox[t * 4 + 1];
        float bx = tbox[t * 4 + 2], by = tbox[t * 4 + 3];
        float areaA = (bx - ax) * (by - ay);
        float iw = fminf(px1, bx) - fmaxf(px0, ax);
        float ih = fminf(py1, by) - fmaxf(py0, ay);
        iw = fmaxf(iw, 0.0f); ih = fmaxf(ih, 0.0f);
        float inter = iw * ih;
        float iou = inter / (areaA + areaB - inter);
        if (iou > best) { best = iou; bidx = t; }   // strict > : first t wins ties (JAX argmax)

        // packed argmax key: iou >= 0 so float bits are order-preserving; ~p -> lowest p wins ties
        unsigned long long pk =
            ((unsigned long long)__float_as_uint(iou) << 32) |
            (unsigned long long)(~(unsigned)p);
#pragma unroll
        for (int off = 16; off > 0; off >>= 1) {
            unsigned long long o = shfl_xor_u64(pk, off);
            if (o > pk) pk = o;
        }
        if (lane == 0) wmax[t * 8 + wv] = pk;
    }
    __syncthreads();

    for (int t = tid; t < T; t += 256) {
        unsigned long long m = wmax[t * 8];
#pragma unroll
        for (int i = 1; i < 8; ++i) {
            unsigned long long v = wmax[t * 8 + i];
            if (v > m) m = v;
        }
        atomicMax(&packed[t], m);
    }

    bto[p] = best;
    bti[p] = bidx;
}

// Phase 2: overrides + encode + smooth-L1 + sigmoid weighting; deterministic per-block partials.
__global__ __launch_bounds__(256) void apl_phase2(
    const float* __restrict__ locs, const float* __restrict__ params,
    const float* __restrict__ truths, int P, int T,
    const unsigned long long* __restrict__ packed,
    const float* __restrict__ bto, const int* __restrict__ bti,
    float* __restrict__ part)
{
    __shared__ __align__(16) float tb[TMAX * 4];
    __shared__ int   q[TMAX];
    __shared__ float s0[256], s1[256];

    int tid = threadIdx.x;
    stage_truths(truths, tb, T, tid);
    for (int t = tid; t < T; t += 256)
        q[t] = (int)(~(unsigned)(packed[t] & 0xffffffffull));
    __syncthreads();

    float a0 = 0.0f, a1 = 0.0f;
    for (int p = blockIdx.x * 256 + tid; p < P; p += gridDim.x * 256) {
        float over = bto[p];
        int   idx  = bti[p];
        int   ov   = -1;
        for (int i = 0; i < T; ++i)
            if (q[i] == p) ov = i;                 // last t wins duplicate scatter
        float xf;
        if (ov >= 0) { idx = ov; xf = 2.5f; }
        else         { xf = (over > 0.4f) ? 1.0f : 0.0f; }

        float x0 = tb[idx * 4 + 0], y0 = tb[idx * 4 + 1];
        float x1 = tb[idx * 4 + 2], y1 = tb[idx * 4 + 3];
        float lx = locs[2 * p + 0], ly = locs[2 * p + 1];
        float w  = params[3 * p + 0], h = params[3 * p + 1];
        float alpha = params[3 * p + 2];

        float gx = fabsf((0.5f * (x0 + x1) - lx) / (0.1f * w));
        float gy = fabsf((0.5f * (y0 + y1) - ly) / (0.1f * h));
        float gw = fabsf(logf((x1 - x0) / w) * 5.0f);   // /0.2
        float gh = fabsf(logf((y1 - y0) / h) * 5.0f);

        float l1 = (gx < 1.0f ? 0.5f * gx * gx : gx - 0.5f)
                 + (gy < 1.0f ? 0.5f * gy * gy : gy - 0.5f)
                 + (gw < 1.0f ? 0.5f * gw * gw : gw - 0.5f)
                 + (gh < 1.0f ? 0.5f * gh * gh : gh - 0.5f);

        float sig = 1.0f / (1.0f + expf(-alpha));
        a0 += sig * (xf * l1 + 1.0f);               // folds BETA*sig into numerator
        a1 += xf;
    }

    s0[tid] = a0; s1[tid] = a1;
    __syncthreads();
#pragma unroll
    for (int s = 128; s > 0; s >>= 1) {
        if (tid < s) { s0[tid] += s0[tid + s]; s1[tid] += s1[tid + s]; }
        __syncthreads();
    }
    if (tid == 0) {
        part[blockIdx.x]       = s0[0];
        part[512 + blockIdx.x] = s1[0];
    }
}

// Final reduction on the matrix pipe: sum 512 numerator + 512 denominator partials
// with V_WMMA_F32_16X16X4_F32 (A = data chunk, B = ones, chained C), then fold the
// 16 rowsums by reinterpreting D VGPR pairs as A matrices (D rows are N-invariant).
__global__ __launch_bounds__(32) void apl_final(const float* __restrict__ part,
                                                float* __restrict__ out)
{
    int l   = threadIdx.x;
    int m   = l & 15;
    int klo = (l >> 4) * 2;

    v2f ones; ones.x = 1.0f; ones.y = 1.0f;
    v8f accN = {}; v8f accD = {};

#pragma unroll
    for (int c = 0; c < 8; ++c) {
        int base = c * 64 + m * 4 + klo;
        v2f a; a.x = part[base];       a.y = part[base + 1];
        v2f b; b.x = part[512 + base]; b.y = part[512 + base + 1];
        accN = __builtin_amdgcn_wmma_f32_16x16x4_f32(false, a, false, ones, (short)0, accN, false, false);
        accD = __builtin_amdgcn_wmma_f32_16x16x4_f32(false, b, false, ones, (short)0, accD, false, false);
    }

    v8f fN = {}; v8f fD = {};
#pragma unroll
    for (int i = 0; i < 4; ++i) {
        v2f an; an.x = accN[2 * i]; an.y = accN[2 * i + 1];
        v2f ad; ad.x = accD[2 * i]; ad.y = accD[2 * i + 1];
        fN = __builtin_amdgcn_wmma_f32_16x16x4_f32(false, an, false, ones, (short)0, fN, false, false);
        fD = __builtin_amdgcn_wmma_f32_16x16x4_f32(false, ad, false, ones, (short)0, fD, false, false);
    }

    if (l == 0) out[0] = fN[0] / fD[0];
}

__global__ void apl_init(unsigned long long* __restrict__ packed, int T) {
    int t = blockIdx.x * blockDim.x + threadIdx.x;
    if (t < T) packed[t] = 0ull;
}

extern "C" void kernel_launch(void* const* d_in, const int* in_sizes, int n_in,
                              void* d_out, int out_size, void* d_ws, size_t ws_size,
                              hipStream_t stream) {
    const float* locs   = (const float*)d_in[0];   // [P,2]
    const float* params = (const float*)d_in[1];   // [P,3]
    const float* truths = (const float*)d_in[2];   // [T,4]
    int P = in_sizes[0] / 2;
    int T = in_sizes[2] / 4;

    char* ws = (char*)d_ws;
    unsigned long long* packed = (unsigned long long*)ws;                 // T * 8B (<=2KB)
    float* bto  = (float*)(ws + 4096);                                    // P floats
    int*   bti  = (int*)  (ws + 4096 + (size_t)P * 4);                    // P ints
    float* part = (float*)(ws + 4096 + (size_t)P * 8);                    // 1024 floats

    apl_init<<<(T + 255) / 256, 256, 0, stream>>>(packed, T);
    apl_phase1<<<(P + 255) / 256, 256, 0, stream>>>(locs, params, truths, P, T, packed, bto, bti);
    apl_phase2<<<512, 256, 0, stream>>>(locs, params, truths, P, T, packed, bto, bti, part);
    apl_final<<<1, 32, 0, stream>>>(part, (float*)d_out);
}